// MLPSCM_61899068670395
// MI455X (gfx1250) — compile-verified
//
#include <hip/hip_runtime.h>
#include <math.h>

// ---------------- problem constants (from reference) ----------------
#define S_LEN 65536
#define NN    256      // num nodes
#define NCA   32       // num causes (roots)
#define LW    16       // nodes per layer
#define NL    14       // layers
#define NF    200      // output features
// ---------------- kernel tiling ----------------
#define WAVES 8        // waves per workgroup, one 16-sample tile each
#define VST   260      // padded LDS stride for vals rows (bank-conflict-free)

typedef __attribute__((ext_vector_type(2))) float v2f;
typedef __attribute__((ext_vector_type(8))) float v8f;

__global__ __launch_bounds__(WAVES * 32) void MLPSCM_scm_fwd_kernel(
    const float* __restrict__ causes,   // [S, 32]
    const float* __restrict__ W,        // [256, 256]
    const float* __restrict__ eps,      // [S, 256]
    const int*   __restrict__ act_ids,  // [256]
    const int*   __restrict__ x_idx,    // [200]
    const int*   __restrict__ y_idx_p,  // [1]
    float*       __restrict__ out)      // [S*200] X then [S] y
{
    extern __shared__ float smem[];
    float* vals = smem;                       // WAVES * 16 * VST
    float* wl   = smem + WAVES * 16 * VST;    // 240 * 16 (staged layer weights)
    int*   xid  = (int*)(wl + 240 * LW);      // 200 staged gather indices

    const int tid  = threadIdx.x;
    const int wave = tid >> 5;
    const int lane = tid & 31;
    const int col  = lane & 15;   // N index within 16x16 tile
    const int rh   = lane >> 4;   // lane-half: selects K pair / M half
    float* mv = vals + wave * 16 * VST;       // this wave's 16x256 vals panel
    const int s0 = (blockIdx.x * WAVES + wave) * 16;

    // stage gather indices once per workgroup
    if (tid < NF) xid[tid] = x_idx[tid];

    // roots: vals[:, 0:32] = clip(causes, -5, 5)   (wave-private panel)
    for (int i = lane; i < 16 * NCA; i += 32) {
        int r = i >> 5, c = i & 31;
        float v = causes[(size_t)(s0 + r) * NCA + c];
        v = fminf(fmaxf(v, -5.0f), 5.0f);
        mv[r * VST + c] = v;
    }

    for (int k = 0; k < NL; ++k) {
        const int cbase = NCA + LW * k;       // first node of this layer
        const int rows  = cbase;              // only earlier nodes can be parents

        __syncthreads();                      // wl free to overwrite
        for (int i = tid; i < rows * LW; i += WAVES * 32)
            wl[i] = W[(size_t)(i >> 4) * NN + cbase + (i & 15)];
        // hint next layer's weight block toward the caches
        if (k + 1 < NL)
            __builtin_prefetch(&W[(size_t)(tid >> 4) * NN + cbase + LW + (tid & 15)], 0, 1);
        __syncthreads();                      // wl ready

        // accumulator C = eps tile (D layout: VGPR j -> row m=j+8*rh, lane -> n=col)
        const int node = cbase + col;
        v8f c;
#pragma unroll
        for (int j = 0; j < 8; ++j)
            c[j] = eps[(size_t)(s0 + j + rh * 8) * NN + node];

        // K loop: Z[16,16] += A[16,4] * B[4,16] over parent chunks
        const float* arow = mv + col * VST + 2 * rh;      // sample=col, K pair base
        const float* bcol = wl + (2 * rh) * LW + col;     // K pair base, N=col
        const int nch = rows >> 2;
        for (int ch = 0; ch < nch; ++ch) {
            v2f a = *(const v2f*)(arow + 4 * ch);         // ds_load_b64, conflict-free
            v2f b;
            b.x = bcol[(4 * ch) * LW];                    // W row 4ch + 2*rh
            b.y = bcol[(4 * ch + 1) * LW];                // W row 4ch + 2*rh + 1
            c = __builtin_amdgcn_wmma_f32_16x16x4_f32(
                    false, a, false, b, (short)0, c, false, false);
        }

        // per-node activation (same node for all 8 regs in a lane)
        const int act = act_ids[node];
#pragma unroll
        for (int j = 0; j < 8; ++j) {
            float z = c[j];
            float r = z;
            if (act == 1)      r = tanhf(z);
            else if (act == 2) r = fmaxf(z, 0.0f);
            else if (act == 3) r = 1.0f / (1.0f + expf(-z));
            c[j] = r;
        }

        // D -> vals panel (transpose via LDS); conflict-free with VST=260
#pragma unroll
        for (int j = 0; j < 8; ++j)
            mv[(j + rh * 8) * VST + node] = c[j];
    }

    // outputs: X = clip(nan_to_num(vals[:, x_idx]), -15, 15); y = nan_to_num(vals[:, y])
    float* outX = out;
    float* outY = out + (size_t)S_LEN * NF;
    for (int i = lane; i < 16 * NF; i += 32) {
        int r = i / NF, c = i - r * NF;
        float v = mv[r * VST + xid[c]];
        v = (v != v) ? 0.0f : v;
        v = fminf(fmaxf(v, -15.0f), 15.0f);
        outX[(size_t)s0 * NF + i] = v;        // == (s0+r)*NF + c, coalesced
    }
    if (lane < 16) {
        float v = mv[lane * VST + y_idx_p[0]];
        v = (v != v) ? 0.0f : v;
        v = fminf(fmaxf(v, -3.402823466e38f), 3.402823466e38f);
        outY[s0 + lane] = v;
    }
}

extern "C" void kernel_launch(void* const* d_in, const int* in_sizes, int n_in,
                              void* d_out, int out_size, void* d_ws, size_t ws_size,
                              hipStream_t stream) {
    const float* causes  = (const float*)d_in[0];
    const float* W       = (const float*)d_in[1];
    const float* eps     = (const float*)d_in[2];
    const int*   act_ids = (const int*)d_in[3];
    const int*   x_idx   = (const int*)d_in[4];
    const int*   y_idx   = (const int*)d_in[5];
    float* out = (float*)d_out;

    const size_t smem_bytes =
        (size_t)(WAVES * 16 * VST + 240 * LW) * sizeof(float) + NF * sizeof(int);
    const int grid = S_LEN / (WAVES * 16);   // 512 workgroups
    MLPSCM_scm_fwd_kernel<<<grid, WAVES * 32, smem_bytes, stream>>>(
        causes, W, eps, act_ids, x_idx, y_idx, out);
}